// SparseEventLinear_9182640079528
// MI455X (gfx1250) — compile-verified
//
#include <hip/hip_runtime.h>
#include <hip/hip_bf16.h>

// ---------------------------------------------------------------------------
// SparseEventLinear on MI455X (gfx1250):
//   out[64,4096] = mask(x) @ W_dense^T + bias,  W densified from COO into d_ws.
//   GEMM on V_WMMA_F32_16X16X4_F32 (fp32 matrix path == reference precision).
//   W (64 MB fp32) is L2-resident (192 MB L2); masked-x precomputed once so
//   the WMMA inner loop is pure global_load_b64 + v_wmma.
// Workspace layout: [0, 64MB)  dense W (fp32 4096x4096)
//                   [64MB, +1MB) masked x (fp32 64x4096)
// ---------------------------------------------------------------------------

#define IN_FEATURES  4096
#define OUT_FEATURES 4096
#define BATCH        64
#define ACTIVE_THRESHOLD 0.01f

typedef __attribute__((ext_vector_type(2))) float v2f;
typedef __attribute__((ext_vector_type(8))) float v8f;

// ---- Kernel 1: zero the dense-W workspace (every call: graph replay must be
// deterministic and the scatter accumulates). ----
__global__ void sel_zero_w(float4* __restrict__ w4, int n4) {
    int i = blockIdx.x * blockDim.x + threadIdx.x;
    if (i < n4) w4[i] = make_float4(0.f, 0.f, 0.f, 0.f);
}

// ---- Kernel 2: event gating, done once: xm = (x > thr) ? x : 0. ----
__global__ void sel_mask_x(const float4* __restrict__ x4,
                           float4* __restrict__ xm4, int n4) {
    int i = blockIdx.x * blockDim.x + threadIdx.x;
    if (i < n4) {
        float4 v = x4[i];
        v.x = (v.x > ACTIVE_THRESHOLD) ? v.x : 0.0f;
        v.y = (v.y > ACTIVE_THRESHOLD) ? v.y : 0.0f;
        v.z = (v.z > ACTIVE_THRESHOLD) ? v.z : 0.0f;
        v.w = (v.w > ACTIVE_THRESHOLD) ? v.w : 0.0f;
        xm4[i] = v;
    }
}

// ---- Kernel 3: COO scatter into dense W (duplicates accumulate -> atomics). ----
__global__ void sel_scatter_w(const int* __restrict__ rows,
                              const int* __restrict__ cols,
                              const float* __restrict__ vals,
                              float* __restrict__ W, int nnz) {
    int k = blockIdx.x * blockDim.x + threadIdx.x;
    if (k < nnz) {
        size_t idx = (size_t)rows[k] * IN_FEATURES + (size_t)cols[k];
        atomicAdd(&W[idx], vals[k]);
    }
}

// ---- Kernel 4: out = xm @ W^T + bias via fp32 WMMA.
// One wave per 16x16 output tile; 1024 tiles (4 M-tiles x 256 N-tiles).
//   A fragment (16x4 f32): lane m = lane&15, k-pair = (lane>>4)*2  -> float2
//   B fragment (4x16 f32): lane n = lane&15, k-pair = (lane>>4)*2  -> float2
//   C/D (16x16 f32, 8 VGPRs): N = lane&15, M = vgpr + (lane>>4)*8
// ----
__global__ __launch_bounds__(256) void sel_wmma_gemm(
        const float* __restrict__ xm,    // [BATCH, IN] pre-masked
        const float* __restrict__ W,     // [OUT, IN] dense
        const float* __restrict__ bias,  // [OUT]
        float* __restrict__ out) {       // [BATCH, OUT]
    const int wave  = blockIdx.x * (blockDim.x >> 5) + (threadIdx.x >> 5);
    const int lane  = threadIdx.x & 31;
    const int mtile = wave & 3;          // 4 M-tiles (BATCH=64)
    const int ntile = wave >> 2;         // 256 N-tiles (OUT=4096)

    const int lm   = lane & 15;
    const int half = lane >> 4;          // which K-pair this lane holds

    const float* xrow = xm + (size_t)(mtile * 16 + lm) * IN_FEATURES + 2 * half;
    const float* wrow = W  + (size_t)(ntile * 16 + lm) * IN_FEATURES + 2 * half;

    v8f acc = {};

    #pragma unroll 8
    for (int k = 0; k < IN_FEATURES; k += 4) {
        v2f a = *(const v2f*)(xrow + k);
        v2f b = *(const v2f*)(wrow + k);
        acc = __builtin_amdgcn_wmma_f32_16x16x4_f32(
            /*neg_a=*/false, a, /*neg_b=*/false, b,
            /*c_mod=*/(short)0, acc, /*reuse_a=*/false, /*reuse_b=*/false);
    }

    const int n = ntile * 16 + lm;
    const float bv = bias[n];
    #pragma unroll
    for (int i = 0; i < 8; ++i) {
        const int m = mtile * 16 + half * 8 + i;
        out[(size_t)m * OUT_FEATURES + n] = acc[i] + bv;
    }
}

extern "C" void kernel_launch(void* const* d_in, const int* in_sizes, int n_in,
                              void* d_out, int out_size, void* d_ws, size_t ws_size,
                              hipStream_t stream) {
    const float* x    = (const float*)d_in[0];
    const int*   rows = (const int*)d_in[1];
    const int*   cols = (const int*)d_in[2];
    const float* vals = (const float*)d_in[3];
    const float* bias = (const float*)d_in[4];
    float* out = (float*)d_out;

    float* W  = (float*)d_ws;                                // 64 MB dense W
    float* xm = W + (size_t)OUT_FEATURES * IN_FEATURES;      // +1 MB masked x
    const int nnz = in_sizes[1];

    // 1) zero dense W (float4 stores)
    const int wN4 = (OUT_FEATURES * IN_FEATURES) / 4;        // 4,194,304
    sel_zero_w<<<(wN4 + 255) / 256, 256, 0, stream>>>((float4*)W, wN4);

    // 2) event gating pass: xm = mask(x)
    const int xN4 = (BATCH * IN_FEATURES) / 4;               // 65,536
    sel_mask_x<<<(xN4 + 255) / 256, 256, 0, stream>>>(
        (const float4*)x, (float4*)xm, xN4);

    // 3) COO -> dense scatter with accumulation
    sel_scatter_w<<<(nnz + 255) / 256, 256, 0, stream>>>(rows, cols, vals, W, nnz);

    // 4) WMMA GEMM + bias: 1024 tiles, 8 waves/block -> 128 blocks x 256 thr
    sel_wmma_gemm<<<128, 256, 0, stream>>>(xm, W, bias, out);
}